// CollisionLoss_89618787598790
// MI455X (gfx1250) — compile-verified
//
#include <hip/hip_runtime.h>

typedef float v2f __attribute__((ext_vector_type(2)));
typedef float v8f __attribute__((ext_vector_type(8)));

#define NPTS    24
#define BATCHES 65536
#define BLOCKS  1024
#define TPB     256
#define WAVES_PER_BLOCK (TPB / 32)
#define TOTAL_WAVES (BLOCKS * WAVES_PER_BLOCK)
#define BATCH_PER_WAVE (BATCHES / TOTAL_WAVES)   // 8, exact

// -4 * log2(e): exp(-4*sq) == exp2(sq * NEG4_LOG2E)
#define NEG4_LOG2E (-5.7707801635558535f)

// static mask: excluded point 0, excluded pair (2,3), no self-pairs, pad >= 24
__host__ __device__ constexpr bool pair_valid(int i, int j) {
    return (i < NPTS) & (j < NPTS) & (i != 0) & (j != 0) & (i != j) &
           !(((i == 2) & (j == 3)) | ((i == 3) & (j == 2)));
}

// compile-time wave32 lane mask of valid pairs for a given tile row k:
// lane L -> i = baseI + k + 8*(L>>4), j = baseJ + (L&15)
__host__ __device__ constexpr unsigned lane_mask(int baseI, int baseJ, int k) {
    unsigned m = 0;
    for (int L = 0; L < 32; ++L) {
        const int i = baseI + k + 8 * (L >> 4);
        const int j = baseJ + (L & 15);
        if (pair_valid(i, j)) m |= (1u << L);
    }
    return m;
}

__global__ void __launch_bounds__(TPB)
collision_main(const float* __restrict__ pos,
               float* __restrict__ sumOut,
               unsigned int* __restrict__ cntOut) {
    const int lane = threadIdx.x & 31;
    const int wave = (int)((blockIdx.x * blockDim.x + threadIdx.x) >> 5);
    const int l15  = lane & 15;
    const int hi   = lane >> 4;          // 0: lanes 0-15, 1: lanes 16-31

    float sum1 = 0.0f, sum2 = 0.0f;              // weight-1 / weight-2 tiles
    unsigned int cnt1 = 0, cnt2 = 0;             // wave-uniform scalar counts

    for (int it = 0; it < BATCH_PER_WAVE; ++it) {
        const int b = wave + it * TOTAL_WAVES;
        const float* base = pos + (size_t)b * (NPTS * 3);

        // Lane owns point p1 = l15 (block 0..15) and p2 = 16+l15 (block 16..23,
        // clamped in-bounds; duplicate rows are killed by the static mask).
        const int p1 = l15;
        int p2 = 16 + l15; p2 = (p2 > NPTS - 1) ? (NPTS - 1) : p2;

        const float x1 = base[p1 * 3 + 0], y1 = base[p1 * 3 + 1], z1 = base[p1 * 3 + 2];
        const float x2 = base[p2 * 3 + 0], y2 = base[p2 * 3 + 1], z2 = base[p2 * 3 + 2];
        const float n1 = x1 * x1 + y1 * y1 + z1 * z1;
        const float n2 = x2 * x2 + y2 * y2 + z2 * z2;

        // A (16x4 f32): row = (x, y, z, |x|^2); v0 = K0|K2, v1 = K1|K3 striping
        v2f a0, a1, bv0, bv1;
        v8f c0, c1;
        a0.x  = hi ? z1 : x1;              a0.y  = hi ? n1 : y1;
        a1.x  = hi ? z2 : x2;              a1.y  = hi ? n2 : y2;
        // B (4x16 f32): col = (-2x, -2y, -2z, 1)
        bv0.x = hi ? (-2.0f * z1) : (-2.0f * x1);   bv0.y = hi ? 1.0f : (-2.0f * y1);
        bv1.x = hi ? (-2.0f * z2) : (-2.0f * x2);   bv1.y = hi ? 1.0f : (-2.0f * y2);
        // C (16x16 f32): C[i][j] = |x_j|^2 ; column index = lane&15 in both halves
        #pragma unroll
        for (int k = 0; k < 8; ++k) { c0[k] = n1; c1[k] = n2; }

        // D = A*B + C => exact squared pairwise distances.
        // Tile (1,0) is the transpose of tile (0,1): skip it, weight (0,1) by 2.
        v8f d00 = __builtin_amdgcn_wmma_f32_16x16x4_f32(false, a0, false, bv0,
                                                        (short)0, c0, false, false);
        v8f d01 = __builtin_amdgcn_wmma_f32_16x16x4_f32(false, a0, false, bv1,
                                                        (short)0, c1, false, false);
        v8f d11 = __builtin_amdgcn_wmma_f32_16x16x4_f32(false, a1, false, bv1,
                                                        (short)0, c1, false, false);

        // pre-scale whole accumulators: exp(-4*sq) = exp2(sq * -4*log2e)
        const v8f t00 = d00 * NEG4_LOG2E;
        const v8f t01 = d01 * NEG4_LOG2E;
        const v8f t11 = d11 * NEG4_LOG2E;

        #pragma unroll
        for (int k = 0; k < 8; ++k) {
            const int im = k + 8 * hi;      // row within a 16-row tile
            const int jn = l15;             // col within a 16-col tile

            {   // tile (0,0): rows 0..15, cols 0..15, weight 1
                const bool cmp = d00[k] < 0.25f;                  // one v_cmp, shared
                cnt1 += (unsigned)__popc(__builtin_amdgcn_ballot_w32(cmp)
                                         & lane_mask(0, 0, k));   // SALU only
                const bool hit = pair_valid(im, jn) & cmp;
                sum1 += hit ? __builtin_amdgcn_exp2f(t00[k]) : 0.0f;
            }
            {   // tile (1,1): rows 16..23, cols 16..23, weight 1
                const bool cmp = d11[k] < 0.25f;
                cnt1 += (unsigned)__popc(__builtin_amdgcn_ballot_w32(cmp)
                                         & lane_mask(16, 16, k));
                const bool hit = pair_valid(16 + im, 16 + jn) & cmp;
                sum1 += hit ? __builtin_amdgcn_exp2f(t11[k]) : 0.0f;
            }
            {   // tile (0,1): rows 0..15, cols 16..23, weight 2 (covers (1,0))
                const bool cmp = d01[k] < 0.25f;
                cnt2 += (unsigned)__popc(__builtin_amdgcn_ballot_w32(cmp)
                                         & lane_mask(0, 16, k));
                const bool hit = pair_valid(im, 16 + jn) & cmp;
                sum2 += hit ? __builtin_amdgcn_exp2f(t01[k]) : 0.0f;
            }
        }
    }

    // wave32 butterfly reduction for the float sum only
    float sumExp = sum1 + 2.0f * sum2;
    #pragma unroll
    for (int off = 16; off > 0; off >>= 1) {
        sumExp += __shfl_xor(sumExp, off, 32);
    }
    if (lane == 0) {
        atomicAdd(sumOut, sumExp);
        atomicAdd(cntOut, cnt1 + 2u * cnt2);   // wave-uniform integer count
    }
}

__global__ void collision_final(const float* __restrict__ sumIn,
                                const unsigned int* __restrict__ cntIn,
                                float* __restrict__ out) {
    const float s = sumIn[0];
    const float c = (float)cntIn[0];
    const float t = (c > 0.0f) ? (s / fmaxf(c, 1.0f)) : 0.0f;
    out[0] = t + 1e-6f;
}

extern "C" void kernel_launch(void* const* d_in, const int* in_sizes, int n_in,
                              void* d_out, int out_size, void* d_ws, size_t ws_size,
                              hipStream_t stream) {
    (void)in_sizes; (void)n_in; (void)out_size; (void)ws_size;
    const float* pos = (const float*)d_in[0];
    float* out = (float*)d_out;
    float* sumP = (float*)d_ws;
    unsigned int* cntP = (unsigned int*)((char*)d_ws + sizeof(float));

    hipMemsetAsync(d_ws, 0, 2 * sizeof(unsigned int), stream);
    collision_main<<<BLOCKS, TPB, 0, stream>>>(pos, sumP, cntP);
    collision_final<<<1, 1, 0, stream>>>(sumP, cntP, out);
}